// net_10548439679178
// MI455X (gfx1250) — compile-verified
//
#include <hip/hip_runtime.h>
#include <hip/hip_bf16.h>

typedef __attribute__((ext_vector_type(2))) float v2f;
typedef __attribute__((ext_vector_type(8))) float v8f;

#define FEAT_PAD 40   // 39 real feature cols + 1 zero pad col
#define HID 256
#define OUT 64
#define LDS_STRIDE 258  // even (8B-aligned v2f reads), avoids 64-bank conflicts

__device__ __forceinline__ void atomic_add_f32(float* p, float v) {
    __hip_atomic_fetch_add(p, v, __ATOMIC_RELAXED, __HIP_MEMORY_SCOPE_AGENT);
}

// ---------------------------------------------------------------------------
// Kernel 1: xl = einsum("nji,nj->ni", gauges, x) into feats[:,0:3];
//           zero feats[:,3:40] (accumulator regions + pad col).
// ---------------------------------------------------------------------------
__global__ __launch_bounds__(256) void gauge_init_kernel(
    const float* __restrict__ x,      // [N,3]
    const float* __restrict__ g,      // [N,3,3]
    float* __restrict__ feats,        // [N,40]
    int nNodes)
{
    int n = blockIdx.x * blockDim.x + threadIdx.x;
    if (n >= nNodes) return;
    float xv0 = x[(size_t)n * 3 + 0];
    float xv1 = x[(size_t)n * 3 + 1];
    float xv2 = x[(size_t)n * 3 + 2];
    const float* gn = g + (size_t)n * 9;
    float* f = feats + (size_t)n * FEAT_PAD;
#pragma unroll
    for (int i = 0; i < 3; ++i) {
        // xl[i] = sum_j gauges[n][j][i] * x[n][j]
        f[i] = gn[0 * 3 + i] * xv0 + gn[1 * 3 + i] * xv1 + gn[2 * 3 + i] * xv2;
    }
#pragma unroll
    for (int i = 3; i < FEAT_PAD; ++i) f[i] = 0.0f;
}

// ---------------------------------------------------------------------------
// Kernel 2: conv layer 1. Reads feats[:,0:3] (xl), scatters 9 f32 atomics
// per edge into feats[:,3:12]. h1[n][c*3+k] += K[k][e] * xl[src[e]][c].
// ---------------------------------------------------------------------------
__global__ __launch_bounds__(256) void aniso_conv1_kernel(
    float* __restrict__ feats,
    const float* __restrict__ kv,     // [3,E]
    const int* __restrict__ src,
    const int* __restrict__ dst,
    int nE)
{
    int e = blockIdx.x * blockDim.x + threadIdx.x;
    if (e >= nE) return;
    size_t s = (size_t)src[e];
    size_t d = (size_t)dst[e];
    const float* fs = feats + s * FEAT_PAD;
    float xs0 = fs[0], xs1 = fs[1], xs2 = fs[2];
    float k0 = kv[e];
    float k1 = kv[(size_t)nE + e];
    float k2 = kv[2 * (size_t)nE + e];
    float* fd = feats + d * FEAT_PAD + 3;
    atomic_add_f32(&fd[0], k0 * xs0);
    atomic_add_f32(&fd[1], k1 * xs0);
    atomic_add_f32(&fd[2], k2 * xs0);
    atomic_add_f32(&fd[3], k0 * xs1);
    atomic_add_f32(&fd[4], k1 * xs1);
    atomic_add_f32(&fd[5], k2 * xs1);
    atomic_add_f32(&fd[6], k0 * xs2);
    atomic_add_f32(&fd[7], k1 * xs2);
    atomic_add_f32(&fd[8], k2 * xs2);
}

// ---------------------------------------------------------------------------
// Kernel 3: conv layer 2. Reads feats[:,3:12] (h1), scatters 27 atomics per
// edge into feats[:,12:39]. h2[n][c*3+k] += K[k][e] * h1[src[e]][c].
// ---------------------------------------------------------------------------
__global__ __launch_bounds__(256) void aniso_conv2_kernel(
    float* __restrict__ feats,
    const float* __restrict__ kv,     // [3,E]
    const int* __restrict__ src,
    const int* __restrict__ dst,
    int nE)
{
    int e = blockIdx.x * blockDim.x + threadIdx.x;
    if (e >= nE) return;
    size_t s = (size_t)src[e];
    size_t d = (size_t)dst[e];
    const float* fs = feats + s * FEAT_PAD + 3;
    float h[9];
#pragma unroll
    for (int c = 0; c < 9; ++c) h[c] = fs[c];
    float k0 = kv[e];
    float k1 = kv[(size_t)nE + e];
    float k2 = kv[2 * (size_t)nE + e];
    float* fd = feats + d * FEAT_PAD + 12;
#pragma unroll
    for (int c = 0; c < 9; ++c) {
        atomic_add_f32(&fd[c * 3 + 0], k0 * h[c]);
        atomic_add_f32(&fd[c * 3 + 1], k1 * h[c]);
        atomic_add_f32(&fd[c * 3 + 2], k2 * h[c]);
    }
}

// ---------------------------------------------------------------------------
// Kernel 4: fused MLP via V_WMMA_F32_16X16X4_F32.
// One wave32 per 16-node tile. Layer 1 (K=40 padded, 16 col-tiles of W1),
// ReLU, stage hmid 16x256 in LDS, Layer 2 (K=256, 4 col-tiles of W2).
// ---------------------------------------------------------------------------
__global__ __launch_bounds__(64) void mlp_wmma_kernel(
    const float* __restrict__ feats,  // [N,40]
    const float* __restrict__ W1,     // [39,256]
    const float* __restrict__ b1,     // [256]
    const float* __restrict__ W2,     // [256,64]
    const float* __restrict__ b2,     // [64]
    float* __restrict__ out,          // [N,64]
    int nNodes, int nTiles)
{
    __shared__ float lds_h[2][16 * LDS_STRIDE];

    const int wave = threadIdx.x >> 5;
    const int lane = threadIdx.x & 31;
    const int tile = blockIdx.x * 2 + wave;
    // Clamp instead of early-return: keeps EXEC all-1s for WMMA and makes
    // __syncthreads() safe even for an overhanging tile.
    const int tileC = tile < nTiles ? tile : (nTiles - 1);
    const int row0 = tileC * 16;
    const int half = lane >> 4;      // 0: lanes 0-15, 1: lanes 16-31
    const int lm   = lane & 15;

    // --- A fragments for layer 1: lane holds A[lm][kk*4 + half*2 + {0,1}] ---
    int rowA = row0 + lm;
    if (rowA >= nNodes) rowA = nNodes - 1;
    const float* frow = feats + (size_t)rowA * FEAT_PAD;
    v2f a1[10];
#pragma unroll
    for (int kk = 0; kk < 10; ++kk) {
        a1[kk] = *(const v2f*)(frow + kk * 4 + half * 2);
    }

    float* hrow = lds_h[wave];

    // --- Layer 1: hmid[16][256] = relu(feats @ W1 + b1) ---
    for (int nt = 0; nt < 16; ++nt) {
        const int col = nt * 16 + lm;
        const float bv = b1[col];
        v8f acc = { bv, bv, bv, bv, bv, bv, bv, bv };
#pragma unroll
        for (int kk = 0; kk < 10; ++kk) {
            const int k0 = kk * 4 + half * 2;
            v2f bfrag;
            bfrag.x = (k0     < 39) ? W1[(size_t)k0       * HID + col] : 0.0f;
            bfrag.y = (k0 + 1 < 39) ? W1[(size_t)(k0 + 1) * HID + col] : 0.0f;
            acc = __builtin_amdgcn_wmma_f32_16x16x4_f32(
                false, a1[kk], false, bfrag, (short)0, acc, false, false);
        }
#pragma unroll
        for (int v = 0; v < 8; ++v) {
            const int m = half * 8 + v;
            const float r = acc[v] > 0.0f ? acc[v] : 0.0f;
            hrow[m * LDS_STRIDE + col] = r;
        }
    }

    __syncthreads();

    // --- Layer 2: emb[16][64] = hmid @ W2 + b2 ---
    for (int nt = 0; nt < 4; ++nt) {
        const int col = nt * 16 + lm;
        const float bv = b2[col];
        v8f acc = { bv, bv, bv, bv, bv, bv, bv, bv };
        for (int kk = 0; kk < 64; ++kk) {
            const int k0 = kk * 4 + half * 2;
            v2f afrag = *(const v2f*)(&hrow[lm * LDS_STRIDE + k0]);
            v2f bfrag;
            bfrag.x = W2[(size_t)k0       * OUT + col];
            bfrag.y = W2[(size_t)(k0 + 1) * OUT + col];
            acc = __builtin_amdgcn_wmma_f32_16x16x4_f32(
                false, afrag, false, bfrag, (short)0, acc, false, false);
        }
        if (tile < nTiles) {
#pragma unroll
            for (int v = 0; v < 8; ++v) {
                const int m = half * 8 + v;
                const int r = row0 + m;
                if (r < nNodes) out[(size_t)r * OUT + col] = acc[v];
            }
        }
    }
}

// ---------------------------------------------------------------------------
// Launch
// Inputs: 0:x [N,3] f32, 1:gauges [N,3,3] f32, 2:kernel_vals [3,E] f32,
//         3:W1 [39,256], 4:b1 [256], 5:W2 [256,64], 6:b2 [64],
//         7:edge_index [2,E] int32
// Output: emb [N,64] f32
// ---------------------------------------------------------------------------
extern "C" void kernel_launch(void* const* d_in, const int* in_sizes, int n_in,
                              void* d_out, int out_size, void* d_ws, size_t ws_size,
                              hipStream_t stream) {
    const float* x      = (const float*)d_in[0];
    const float* gauges = (const float*)d_in[1];
    const float* kv     = (const float*)d_in[2];
    const float* W1     = (const float*)d_in[3];
    const float* b1     = (const float*)d_in[4];
    const float* W2     = (const float*)d_in[5];
    const float* b2     = (const float*)d_in[6];
    const int*   eidx   = (const int*)d_in[7];

    const int nNodes = in_sizes[0] / 3;
    const int nE     = in_sizes[7] / 2;
    const int* src = eidx;
    const int* dst = eidx + nE;

    float* feats = (float*)d_ws;           // [N,40] packed padded features
    float* outp  = (float*)d_out;          // [N,64]

    // 1) gauge projection + zero-init of accumulator columns
    {
        int blocks = (nNodes + 255) / 256;
        gauge_init_kernel<<<blocks, 256, 0, stream>>>(x, gauges, feats, nNodes);
    }
    // 2) conv layer 1 (scatter-add into feats[:,3:12])
    {
        int blocks = (nE + 255) / 256;
        aniso_conv1_kernel<<<blocks, 256, 0, stream>>>(feats, kv, src, dst, nE);
    }
    // 3) conv layer 2 (scatter-add into feats[:,12:39])
    {
        int blocks = (nE + 255) / 256;
        aniso_conv2_kernel<<<blocks, 256, 0, stream>>>(feats, kv, src, dst, nE);
    }
    // 4) fused MLP with f32 WMMA (one wave per 16-node tile, 2 waves/block)
    {
        int nTiles = (nNodes + 15) / 16;
        int blocks = (nTiles + 1) / 2;
        mlp_wmma_kernel<<<blocks, 64, 0, stream>>>(feats, W1, b1, W2, b2, outp,
                                                   nNodes, nTiles);
    }
}